// CrossAttentionGNNConv_81561428951591
// MI455X (gfx1250) — compile-verified
//
#include <hip/hip_runtime.h>
#include <hip/hip_bf16.h>

typedef __attribute__((ext_vector_type(2))) float v2f;
typedef __attribute__((ext_vector_type(8))) float v8f;

#define D 128
#define INV_SQRT_D 0.08838834764831845f   // 1/sqrt(128)

// ---------------- order-preserving float<->uint key for atomic max ----------
__device__ __forceinline__ unsigned fkey(float f) {
    unsigned u = __float_as_uint(f);
    return (u & 0x80000000u) ? ~u : (u | 0x80000000u);
}
__device__ __forceinline__ float funkey(unsigned k) {
    return __uint_as_float((k & 0x80000000u) ? (k ^ 0x80000000u) : ~k);
}

// ---------------- K0: zero outputs + node max/denominator scratch -----------
__global__ void k_init(float* __restrict__ out, unsigned* __restrict__ tail,
                       long outWords, long tailWords) {
    long i = (long)blockIdx.x * blockDim.x + threadIdx.x;
    if (i < outWords)  out[i]  = 0.0f;
    if (i < tailWords) tail[i] = 0u;
}

// ---------------- K1: six node projections via V_WMMA_F32_16X16X4_F32 -------
// One block = one 16-row node tile. 8 waves sweep 48 (matrix, col-tile) tasks.
// out[n,d] = sum_k in[n,k] * W[d,k] (+ bias[d])
__global__ __launch_bounds__(256)
void k_node_proj(const float* __restrict__ x, const float* __restrict__ t,
                 const float* __restrict__ Wx,  const float* __restrict__ Wt,
                 const float* __restrict__ Qaw, const float* __restrict__ Qab,
                 const float* __restrict__ Kaw, const float* __restrict__ Kab,
                 const float* __restrict__ Qbw, const float* __restrict__ Qbb,
                 const float* __restrict__ Kbw, const float* __restrict__ Kbb,
                 float* __restrict__ qa, float* __restrict__ ka,
                 float* __restrict__ wtn,
                 float* __restrict__ qb, float* __restrict__ kb,
                 float* __restrict__ wxn, int N) {
    __shared__ float tileT[16 * D];
    __shared__ float tileX[16 * D];

    const int tid      = threadIdx.x;
    const int row0     = blockIdx.x * 16;
    const bool fulltile = (row0 + 16 <= N);   // uniform: true except last tile

    // cooperative stage of the 16x128 t/x tiles into LDS (float4 granules)
    if (fulltile) {
        for (int i = tid; i < 16 * (D / 4); i += 256) {
            int gr = row0 + (i >> 5);
            reinterpret_cast<float4*>(tileT)[i] =
                reinterpret_cast<const float4*>(t + (size_t)gr * D)[i & 31];
            reinterpret_cast<float4*>(tileX)[i] =
                reinterpret_cast<const float4*>(x + (size_t)gr * D)[i & 31];
        }
    } else {
        for (int i = tid; i < 16 * (D / 4); i += 256) {
            int gr = row0 + (i >> 5); if (gr >= N) gr = N - 1;
            reinterpret_cast<float4*>(tileT)[i] =
                reinterpret_cast<const float4*>(t + (size_t)gr * D)[i & 31];
            reinterpret_cast<float4*>(tileX)[i] =
                reinterpret_cast<const float4*>(x + (size_t)gr * D)[i & 31];
        }
    }
    __syncthreads();

    const int wave = tid >> 5;
    const int lane = tid & 31;
    const int lo   = lane & 15;   // M index (A) / N index (B,C)
    const int hi   = lane >> 4;   // K-half selector

    for (int task = wave; task < 48; task += 8) {
        const int m = task >> 3;          // which projection
        const int cblk = task & 7;        // 16-col tile
        const float* A;  const float* W;  const float* bias;  float* O;
        switch (m) {
            case 0: A = tileT; W = Qaw; bias = Qab; O = qa;  break;
            case 1: A = tileT; W = Kaw; bias = Kab; O = ka;  break;
            case 2: A = tileT; W = Wt;  bias = 0;   O = wtn; break;
            case 3: A = tileX; W = Qbw; bias = Qbb; O = qb;  break;
            case 4: A = tileX; W = Kbw; bias = Kbb; O = kb;  break;
            default:A = tileX; W = Wx;  bias = 0;   O = wxn; break;
        }
        const int colg = cblk * 16 + lo;
        // ISA A-layout 16x4 f32: lane(0..15)=M, vgpr0/1 = K,K+1; hi half = K+2,K+3
        const float* arow = A + lo * D + 2 * hi;
        // ISA B-layout 4x16: lane = N, vgpr0/1 = K,K+1; hi half = K+2,K+3
        const float* wrow = W + (size_t)colg * D + 2 * hi;

        v8f acc = {};
        #pragma unroll 4
        for (int k = 0; k < D; k += 4) {
            v2f a = *reinterpret_cast<const v2f*>(arow + k);
            v2f b = *reinterpret_cast<const v2f*>(wrow + k);
            acc = __builtin_amdgcn_wmma_f32_16x16x4_f32(
                false, a, false, b, (short)0, acc, false, false);
        }

        const float bv = bias ? bias[colg] : 0.0f;
        // C layout: vgpr r -> M=r (lanes 0-15) / M=r+8 (lanes 16-31), N=lo
        float* obase = O + (size_t)(row0 + 8 * hi) * D + colg;
        if (fulltile) {
            // straight-line stores, no exec-mask juggling (common path)
            #pragma unroll
            for (int r = 0; r < 8; ++r) obase[(size_t)r * D] = acc[r] + bv;
        } else {
            #pragma unroll
            for (int r = 0; r < 8; ++r) {
                if (row0 + 8 * hi + r < N) obase[(size_t)r * D] = acc[r] + bv;
            }
        }
    }
}

// ---------------- K2: wave-per-edge score dots + segment max ----------------
__global__ __launch_bounds__(256)
void k_edge_scores(const int* __restrict__ ei,
                   const float* __restrict__ qa, const float* __restrict__ ka,
                   const float* __restrict__ qb, const float* __restrict__ kb,
                   float* __restrict__ sa, float* __restrict__ sb,
                   unsigned* __restrict__ ama, unsigned* __restrict__ amb,
                   int E) {
    const int lane = threadIdx.x & 31;
    const int gw   = (int)((blockIdx.x * blockDim.x + threadIdx.x) >> 5);
    const int nw   = (int)((gridDim.x * blockDim.x) >> 5);
    for (int e = gw; e < E; e += nw) {
        const int r = ei[e];        // dest
        const int c = ei[E + e];    // src
        float4 q1 = reinterpret_cast<const float4*>(qa + (size_t)r * D)[lane];
        float4 k1 = reinterpret_cast<const float4*>(ka + (size_t)c * D)[lane];
        float4 q2 = reinterpret_cast<const float4*>(qb + (size_t)r * D)[lane];
        float4 k2 = reinterpret_cast<const float4*>(kb + (size_t)c * D)[lane];
        float pa = q1.x*k1.x + q1.y*k1.y + q1.z*k1.z + q1.w*k1.w;
        float pb = q2.x*k2.x + q2.y*k2.y + q2.z*k2.z + q2.w*k2.w;
        #pragma unroll
        for (int off = 16; off >= 1; off >>= 1) {
            pa += __shfl_xor(pa, off, 32);
            pb += __shfl_xor(pb, off, 32);
        }
        if (lane == 0) {
            float va = pa * INV_SQRT_D, vb = pb * INV_SQRT_D;
            sa[e] = va; sb[e] = vb;
            atomicMax(&ama[r], fkey(va));
            atomicMax(&amb[r], fkey(vb));
        }
    }
}

// ---------------- K3: exp(score - max), accumulate denominators -------------
__global__ void k_edge_exp(const int* __restrict__ ei,
                           float* __restrict__ sa, float* __restrict__ sb,
                           const unsigned* __restrict__ ama,
                           const unsigned* __restrict__ amb,
                           float* __restrict__ dena, float* __restrict__ denb,
                           int E) {
    int e = blockIdx.x * blockDim.x + threadIdx.x;
    if (e >= E) return;
    const int r = ei[e];
    float ea = expf(sa[e] - funkey(ama[r]));
    float eb = expf(sb[e] - funkey(amb[r]));
    sa[e] = ea; sb[e] = eb;
    atomicAdd(&dena[r], ea);
    atomicAdd(&denb[r], eb);
}

// ---------------- K4: wave-per-edge weighted message scatter ----------------
__global__ __launch_bounds__(256)
void k_edge_scatter(const int* __restrict__ ei,
                    const float* __restrict__ sa, const float* __restrict__ sb,
                    const float* __restrict__ dena, const float* __restrict__ denb,
                    const float* __restrict__ wtn, const float* __restrict__ wxn,
                    float* __restrict__ out_x, float* __restrict__ out_t,
                    int E) {
    const int lane = threadIdx.x & 31;
    const int gw   = (int)((blockIdx.x * blockDim.x + threadIdx.x) >> 5);
    const int nw   = (int)((gridDim.x * blockDim.x) >> 5);
    for (int e = gw; e < E; e += nw) {
        const int r = ei[e];
        const int c = ei[E + e];
        const float alpha = sa[e] / dena[r];
        const float beta  = sb[e] / denb[r];
        float4 mt = reinterpret_cast<const float4*>(wtn + (size_t)c * D)[lane];
        float4 mx = reinterpret_cast<const float4*>(wxn + (size_t)c * D)[lane];
        float* ot = out_t + (size_t)r * D + lane * 4;
        float* ox = out_x + (size_t)r * D + lane * 4;
        atomicAdd(ot + 0, alpha * mt.x);
        atomicAdd(ot + 1, alpha * mt.y);
        atomicAdd(ot + 2, alpha * mt.z);
        atomicAdd(ot + 3, alpha * mt.w);
        atomicAdd(ox + 0, beta * mx.x);
        atomicAdd(ox + 1, beta * mx.y);
        atomicAdd(ox + 2, beta * mx.z);
        atomicAdd(ox + 3, beta * mx.w);
    }
}

extern "C" void kernel_launch(void* const* d_in, const int* in_sizes, int n_in,
                              void* d_out, int out_size, void* d_ws, size_t ws_size,
                              hipStream_t stream) {
    (void)n_in; (void)out_size; (void)ws_size;
    const float* x   = (const float*)d_in[0];
    const float* t   = (const float*)d_in[1];
    const int*   ei  = (const int*)  d_in[2];
    const float* Wx  = (const float*)d_in[3];
    const float* Wt  = (const float*)d_in[4];
    const float* Qaw = (const float*)d_in[5];
    const float* Qab = (const float*)d_in[6];
    const float* Kaw = (const float*)d_in[7];
    const float* Kab = (const float*)d_in[8];
    const float* Qbw = (const float*)d_in[9];
    const float* Qbb = (const float*)d_in[10];
    const float* Kbw = (const float*)d_in[11];
    const float* Kbb = (const float*)d_in[12];

    const int N = in_sizes[0] / D;
    const int E = in_sizes[2] / 2;
    const size_t ND = (size_t)N * D;

    float* out_x = (float*)d_out;
    float* out_t = out_x + ND;

    // workspace carve-up (~158 MB)
    float* qa  = (float*)d_ws;
    float* ka  = qa + ND;
    float* qb  = ka + ND;
    float* kb  = qb + ND;
    float* wtn = kb + ND;
    float* wxn = wtn + ND;
    float* sa  = wxn + ND;
    float* sb  = sa + E;
    unsigned* ama  = (unsigned*)(sb + E);
    unsigned* amb  = ama + N;
    float*    dena = (float*)(amb + N);
    float*    denb = dena + N;

    // K0: zero out + (amax|den) tail (4N words, contiguous after ama)
    {
        long outWords = 2L * (long)ND;
        long tailWords = 4L * (long)N;
        long mx = outWords > tailWords ? outWords : tailWords;
        int blocks = (int)((mx + 255) / 256);
        k_init<<<blocks, 256, 0, stream>>>(out_x, ama, outWords, tailWords);
    }
    // K1: node projections (WMMA f32)
    {
        int tiles = (N + 15) / 16;
        k_node_proj<<<tiles, 256, 0, stream>>>(x, t, Wx, Wt, Qaw, Qab, Kaw, Kab,
                                               Qbw, Qbb, Kbw, Kbb,
                                               qa, ka, wtn, qb, kb, wxn, N);
    }
    // K2: edge scores + segment max
    {
        int blocks = (E + 7) / 8;            // 8 waves/block, 1 edge/wave
        k_edge_scores<<<blocks, 256, 0, stream>>>(ei, qa, ka, qb, kb,
                                                  sa, sb, ama, amb, E);
    }
    // K3: exp + denominators
    {
        int blocks = (E + 255) / 256;
        k_edge_exp<<<blocks, 256, 0, stream>>>(ei, sa, sb, ama, amb,
                                               dena, denb, E);
    }
    // K4: message scatter
    {
        int blocks = (E + 7) / 8;
        k_edge_scatter<<<blocks, 256, 0, stream>>>(ei, sa, sb, dena, denb,
                                                   wtn, wxn, out_x, out_t, E);
    }
}